// AttentionVTP_57320633533122
// MI455X (gfx1250) — compile-verified
//
#include <hip/hip_runtime.h>
#include <hip/hip_bf16.h>

typedef __attribute__((ext_vector_type(2))) float v2f;
typedef __attribute__((ext_vector_type(4))) float v4f;
typedef __attribute__((ext_vector_type(8))) float v8f;

#define NPATCH   1024
#define DIMF     512
#define NHEADS   8
#define DHEAD    64
#define NBATCH   8
#define KD       409            // pruned input channels
#define KDP      412            // KD padded to multiple of 4 (zero-filled)
#define KH2      51             // pruned per-head channels
#define QKVN     1536           // 3 * heads * dim_head
#define KO       (NHEADS * KH2) // 408
#define BH       (NBATCH * NHEADS)
#define YELEMS   (NBATCH * NPATCH * DIMF)
#define ATTN_SCALE 0.125f       // 64^-0.5

// ---- f32 WMMA 16x16x4 wrapper (D = A(16x4) * B(4x16) + C) ----
static __device__ __forceinline__ v8f wmma4(v2f a, v2f b, v8f c) {
  return __builtin_amdgcn_wmma_f32_16x16x4_f32(false, a, false, b, (short)0, c,
                                               false, false);
}

static __device__ __forceinline__ v8f v8zero() {
  v8f z = {0.f, 0.f, 0.f, 0.f, 0.f, 0.f, 0.f, 0.f};
  return z;
}

// ======================= pruning / scoring kernels ==========================

// s1[b][c] = sum_n x[b][n][c] * W[n] + bias  (grid=8, block=512)
__global__ void scores1_kernel(const float* __restrict__ x,
                               const float* __restrict__ w,
                               const float* __restrict__ bb,
                               float* __restrict__ s1) {
  int b = blockIdx.x, c = threadIdx.x;
  const float* xb = x + (size_t)b * NPATCH * DIMF + c;
  float acc = 0.f;
  for (int n = 0; n < NPATCH; ++n) acc += xb[(size_t)n * DIMF] * w[n];
  s1[b * DIMF + c] = acc + bb[0];
}

// Order-preserving top-K via stable rank. blockDim == N (N <= 512).
__global__ void topk_keep_kernel(const float* __restrict__ scores, int N, int K,
                                 int* __restrict__ idx_out) {
  __shared__ float sv[512];
  __shared__ int keep[512];
  int g = blockIdx.x, i = threadIdx.x;
  sv[i] = scores[g * N + i];
  __syncthreads();
  float si = sv[i];
  int r = 0;
  for (int j = 0; j < N; ++j) {
    float sj = sv[j];
    r += (sj > si) || (sj == si && j < i);
  }
  keep[i] = (r < K) ? 1 : 0;
  __syncthreads();
  if (keep[i]) {
    int p = 0;
    for (int j = 0; j < i; ++j) p += keep[j];
    idx_out[g * K + p] = i;
  }
}

// xp[b][n][j] = x[b][n][idx1[b][j]], zero-padded to KDP columns
__global__ void gather_xp_kernel(const float* __restrict__ x,
                                 const int* __restrict__ idx1,
                                 float* __restrict__ xp) {
  int t = blockIdx.x * 256 + threadIdx.x;  // over 8192 * KDP
  int j = t % KDP;
  int bn = t / KDP;  // b*1024+n
  int b = bn >> 10;
  xp[t] = (j < KD) ? x[(size_t)bn * DIMF + idx1[b * KD + j]] : 0.f;
}

// Wp: K-pair interleaved, zero-padded W_qkv: Wp[(r>>1)*QKVN*2 + c*2 + (r&1)]
__global__ void wpad_kernel(const float* __restrict__ W, float* __restrict__ Wp) {
  int t = blockIdx.x * 256 + threadIdx.x;  // over KDP*QKVN
  int r = t / QKVN, c = t % QKVN;
  float val = (r < KD) ? W[t] : 0.f;
  Wp[(((size_t)(r >> 1)) * QKVN + c) * 2 + (r & 1)] = val;
}

// Wop: K-pair interleaved W_out[408][512]
__global__ void wopad_kernel(const float* __restrict__ W, float* __restrict__ Wp) {
  int t = blockIdx.x * 256 + threadIdx.x;  // over KO*DIMF
  int r = t / DIMF, c = t % DIMF;
  Wp[(((size_t)(r >> 1)) * DIMF + c) * 2 + (r & 1)] = W[t];
}

// ========================= QKV GEMM (WMMA f32) ==============================
// C[8192 x 1536] = xp[8192 x 412pad] @ Wp; one wave computes a 16x64 strip
// (A-fragment reused across 4 N-subtiles). A and B fragments are both v2f
// (b64) loads thanks to the K-pair interleaved B layout. Q,K scattered to
// [b,h,n,64]; V scattered to j-pair interleaved [bh][n/2][d][2] for pass 2.
__global__ void qkv_gemm_kernel(const float* __restrict__ A,
                                const float* __restrict__ B,
                                float* __restrict__ Qm, float* __restrict__ Km,
                                float* __restrict__ Vt) {
  int wid = threadIdx.x >> 5, lane = threadIdx.x & 31;
  int task = blockIdx.x * 8 + wid;
  const int tilesN = QKVN / 64;  // 24
  int tm = task / tilesN, tn = task % tilesN;
  int m = lane & 15, kh = lane >> 4;
  int row = tm * 16 + m;
  int col0 = tn * 64 + m;
  v8f acc[4];
#pragma unroll
  for (int dt = 0; dt < 4; ++dt) acc[dt] = v8zero();

  const float* arow = A + (size_t)row * KDP;
  for (int kk = 0; kk < KDP; kk += 4) {
    int ka = kk + 2 * kh;                 // even
    v2f a = *(const v2f*)(arow + ka);
    const float* b0 = B + (((size_t)(ka >> 1)) * QKVN + col0) * 2;
#pragma unroll
    for (int dt = 0; dt < 4; ++dt) {
      v2f bf = *(const v2f*)(b0 + dt * 32);  // rows ka, ka+1 interleaved
      acc[dt] = wmma4(a, bf, acc[dt]);
    }
  }
  // 64-wide tiles are aligned: q/k/v selector and head are wave constants
  int which = tn >> 3;  // 0=q,1=k,2=v
  int h = tn & 7;
  if (which == 2) {
#pragma unroll
    for (int dt = 0; dt < 4; ++dt) {
      int d = dt * 16 + m;
#pragma unroll
      for (int v = 0; v < 8; ++v) {
        int r = tm * 16 + v + 8 * kh;  // b*1024+n
        int b = r >> 10, n = r & 1023;
        Vt[(((size_t)(b * NHEADS + h)) * (NPATCH / 2) + (n >> 1)) * (2 * DHEAD) +
           d * 2 + (n & 1)] = acc[dt][v];
      }
    }
  } else {
    float* dst = (which == 0) ? Qm : Km;
#pragma unroll
    for (int dt = 0; dt < 4; ++dt) {
      int d = dt * 16 + m;
#pragma unroll
      for (int v = 0; v < 8; ++v) {
        int r = tm * 16 + v + 8 * kh;
        int b = r >> 10, n = r & 1023;
        dst[(((size_t)(b * NHEADS + h)) * NPATCH + n) * DHEAD + d] = acc[dt][v];
      }
    }
  }
}

// ================== RPE bias plane: rpb + 0.01*exp(-f*dis) ==================
__global__ void bias_kernel(const float* __restrict__ rpb,
                            const float* __restrict__ sita,
                            const int* __restrict__ rel_index,
                            const float* __restrict__ dis,
                            float* __restrict__ biasA) {
  int ij = blockIdx.x * 256 + threadIdx.x;
  int ri = rel_index[ij];
  float dd = dis[ij];
#pragma unroll
  for (int h = 0; h < NHEADS; ++h) {
    float th = sita[h];
    float f = 1.f / (2.f * th * th + 1e-10f);
    biasA[((size_t)h << 20) + ij] = rpb[ri * NHEADS + h] + 0.01f * __expf(-f * dd);
  }
}

// =================== attention pass 1: row maxes (m0, m1) ===================
__global__ void attn_pass1_kernel(const float* __restrict__ Qm,
                                  const float* __restrict__ Km,
                                  const float* __restrict__ biasA,
                                  float* __restrict__ m0s,
                                  float* __restrict__ m1s) {
  int wid = threadIdx.x >> 5, lane = threadIdx.x & 31;
  int task = blockIdx.x * 8 + wid;
  int bh = task >> 6, it = task & 63;
  int i0 = it * 16;
  int m = lane & 15, kh = lane >> 4;
  int h = bh & (NHEADS - 1);
  const float* q = Qm + (size_t)bh * NPATCH * DHEAD;
  const float* k = Km + (size_t)bh * NPATCH * DHEAD;

  v2f qa[16];
#pragma unroll
  for (int t = 0; t < 16; ++t)
    qa[t] = *(const v2f*)(q + (size_t)(i0 + m) * DHEAD + t * 4 + 2 * kh);

  float mx0[8], mx1[8];
#pragma unroll
  for (int v = 0; v < 8; ++v) { mx0[v] = -3.4e38f; mx1[v] = -3.4e38f; }

  for (int j0 = 0; j0 < NPATCH; j0 += 16) {
    const float* krow = k + (size_t)(j0 + m) * DHEAD + 2 * kh;
    v8f s = v8zero();
#pragma unroll
    for (int t = 0; t < 16; ++t) {
      v2f bf = *(const v2f*)(krow + t * 4);
      s = wmma4(qa[t], bf, s);
    }
#pragma unroll
    for (int v = 0; v < 8; ++v) {
      float d0 = s[v] * ATTN_SCALE;
      float bi = biasA[((size_t)h << 20) +
                       ((size_t)(i0 + v + 8 * kh) << 10) + j0 + m];
      float d1 = d0 + bi;
      mx0[v] = fmaxf(mx0[v], d0);
      mx1[v] = fmaxf(mx1[v], d1);
    }
  }
#pragma unroll
  for (int v = 0; v < 8; ++v) {
#pragma unroll
    for (int off = 1; off < 16; off <<= 1) {
      mx0[v] = fmaxf(mx0[v], __shfl_xor(mx0[v], off, 32));
      mx1[v] = fmaxf(mx1[v], __shfl_xor(mx1[v], off, 32));
    }
  }
  if (m == 0) {
#pragma unroll
    for (int v = 0; v < 8; ++v) {
      int r = bh * NPATCH + i0 + v + 8 * kh;
      m0s[r] = mx0[v];
      m1s[r] = mx1[v];
    }
  }
}

// ====== attention pass 2: exp/sums, attn0 out, P@V via WMMA through LDS =====
__global__ void attn_pass2_kernel(const float* __restrict__ Qm,
                                  const float* __restrict__ Km,
                                  const float* __restrict__ Vt,
                                  const float* __restrict__ biasA,
                                  const float* __restrict__ m0s,
                                  const float* __restrict__ m1s,
                                  float* __restrict__ attn0,
                                  float* __restrict__ s0s,
                                  float* __restrict__ out_s) {
  __shared__ float plds[8][16][18];  // stride 18: conflict-free + 8B aligned
  int wid = threadIdx.x >> 5, lane = threadIdx.x & 31;
  int task = blockIdx.x * 8 + wid;
  int bh = task >> 6, it = task & 63;
  int i0 = it * 16;
  int m = lane & 15, kh = lane >> 4;
  int h = bh & (NHEADS - 1);
  const float* q = Qm + (size_t)bh * NPATCH * DHEAD;
  const float* k = Km + (size_t)bh * NPATCH * DHEAD;
  const float* vt = Vt + (size_t)bh * NPATCH * DHEAD;  // [n/2][d][2]

  v2f qa[16];
#pragma unroll
  for (int t = 0; t < 16; ++t)
    qa[t] = *(const v2f*)(q + (size_t)(i0 + m) * DHEAD + t * 4 + 2 * kh);

  float mx0[8], mx1[8], sum0[8], sum1[8];
#pragma unroll
  for (int v = 0; v < 8; ++v) {
    int r = bh * NPATCH + i0 + v + 8 * kh;
    mx0[v] = m0s[r];
    mx1[v] = m1s[r];
    sum0[v] = 0.f;
    sum1[v] = 0.f;
  }
  v8f acc[4];
#pragma unroll
  for (int dt = 0; dt < 4; ++dt) acc[dt] = v8zero();

  for (int j0 = 0; j0 < NPATCH; j0 += 16) {  // uniform trip count: barriers OK
    const float* krow = k + (size_t)(j0 + m) * DHEAD + 2 * kh;
    v8f s = v8zero();
#pragma unroll
    for (int t = 0; t < 16; ++t) {
      v2f bf = *(const v2f*)(krow + t * 4);
      s = wmma4(qa[t], bf, s);
    }
#pragma unroll
    for (int v = 0; v < 8; ++v) {
      float d0 = s[v] * ATTN_SCALE;
      float bi = biasA[((size_t)h << 20) +
                       ((size_t)(i0 + v + 8 * kh) << 10) + j0 + m];
      float d1 = d0 + bi;
      float e0 = __expf(d0 - mx0[v]);
      float e1 = __expf(d1 - mx1[v]);
      sum0[v] += e0;
      sum1[v] += e1;
      // unnormalized softmax(dots0): normalized in attn0_norm_kernel
      attn0[((size_t)(bh * NPATCH + i0 + v + 8 * kh) << 10) + j0 + m] = e0;
      plds[wid][v + 8 * kh][m] = e1;  // P tile for WMMA A-fragments
    }
    __syncthreads();
    // out(16x64) += P(16x16) @ V(16x64); V rows pair-interleaved -> v2f loads
#pragma unroll
    for (int kk2 = 0; kk2 < 16; kk2 += 4) {
      v2f a = *(const v2f*)(&plds[wid][m][kk2 + 2 * kh]);
      const float* vrow =
          vt + (size_t)((j0 + kk2) / 2 + kh) * (2 * DHEAD) + m * 2;
#pragma unroll
      for (int dt = 0; dt < 4; ++dt) {
        v2f bf = *(const v2f*)(vrow + dt * 32);  // V[j][d], V[j+1][d]
        acc[dt] = wmma4(a, bf, acc[dt]);
      }
    }
    __syncthreads();
  }
#pragma unroll
  for (int v = 0; v < 8; ++v) {
#pragma unroll
    for (int off = 1; off < 16; off <<= 1) {
      sum0[v] += __shfl_xor(sum0[v], off, 32);
      sum1[v] += __shfl_xor(sum1[v], off, 32);
    }
  }
  if (m == 0) {
#pragma unroll
    for (int v = 0; v < 8; ++v)
      s0s[bh * NPATCH + i0 + v + 8 * kh] = sum0[v];
  }
  float inv[8];
#pragma unroll
  for (int v = 0; v < 8; ++v) inv[v] = 1.f / sum1[v];
#pragma unroll
  for (int dt = 0; dt < 4; ++dt) {
#pragma unroll
    for (int v = 0; v < 8; ++v) {
      out_s[((size_t)(bh * NPATCH + i0 + v + 8 * kh)) * DHEAD + dt * 16 + m] =
          acc[dt][v] * inv[v];
    }
  }
}

// normalize the streamed attn0 output by per-row sums (vectorized x4)
__global__ void attn0_norm_kernel(float* __restrict__ attn0,
                                  const float* __restrict__ s0s) {
  int t = blockIdx.x * 256 + threadIdx.x;  // over 16M vec4 groups
  int row = t >> 8;                        // (t*4) >> 10
  float inv = 1.f / s0s[row];
  v4f* p = (v4f*)attn0 + t;
  v4f e = *p;
  e.x *= inv; e.y *= inv; e.z *= inv; e.w *= inv;
  *p = e;
}

// ===================== output pruning + final projection ====================

__global__ void scores2_kernel(const float* __restrict__ out_s,
                               const float* __restrict__ w,
                               const float* __restrict__ bb,
                               float* __restrict__ s2) {
  int bh = blockIdx.x, d = threadIdx.x;
  const float* p = out_s + (size_t)bh * NPATCH * DHEAD + d;
  float acc = 0.f;
  for (int n = 0; n < NPATCH; ++n) acc += p[(size_t)n * DHEAD] * w[n];
  s2[bh * DHEAD + d] = acc + bb[0];
}

// o2[b][n][h*51+jj] = out_s[b][h][n][idx2[b][h][jj]]
__global__ void gather_out2_kernel(const float* __restrict__ out_s,
                                   const int* __restrict__ idx2,
                                   float* __restrict__ o2) {
  int t = blockIdx.x * 256 + threadIdx.x;
  int c = t % KO;
  int bn = t / KO;
  int b = bn >> 10, n = bn & 1023;
  int h = c / KH2, jj = c % KH2;
  int d = idx2[(b * NHEADS + h) * KH2 + jj];
  o2[t] = out_s[(((size_t)(b * NHEADS + h)) * NPATCH + n) * DHEAD + d];
}

// y[8192 x 512] = o2[8192 x 408] @ Wop + b_out; 16x64 strip per wave;
// all fragment loads are v2f (b64).
__global__ void out_gemm_kernel(const float* __restrict__ A,
                                const float* __restrict__ B,
                                const float* __restrict__ bias,
                                float* __restrict__ y) {
  int wid = threadIdx.x >> 5, lane = threadIdx.x & 31;
  int task = blockIdx.x * 8 + wid;
  const int tilesN = DIMF / 64;  // 8
  int tm = task / tilesN, tn = task % tilesN;
  int m = lane & 15, kh = lane >> 4;
  int row = tm * 16 + m;
  int col0 = tn * 64 + m;
  v8f acc[4];
#pragma unroll
  for (int dt = 0; dt < 4; ++dt) acc[dt] = v8zero();

  const float* arow = A + (size_t)row * KO;
  for (int kk = 0; kk < KO; kk += 4) {
    int ka = kk + 2 * kh;  // even
    v2f a = *(const v2f*)(arow + ka);
    const float* b0 = B + (((size_t)(ka >> 1)) * DIMF + col0) * 2;
#pragma unroll
    for (int dt = 0; dt < 4; ++dt) {
      v2f bf = *(const v2f*)(b0 + dt * 32);
      acc[dt] = wmma4(a, bf, acc[dt]);
    }
  }
#pragma unroll
  for (int dt = 0; dt < 4; ++dt) {
    float badd = bias[col0 + dt * 16];
#pragma unroll
    for (int v = 0; v < 8; ++v) {
      int r = tm * 16 + v + 8 * kh;
      y[(size_t)r * DIMF + col0 + dt * 16] = acc[dt][v] + badd;
    }
  }
}

// ================================ launcher ==================================
extern "C" void kernel_launch(void* const* d_in, const int* in_sizes, int n_in,
                              void* d_out, int out_size, void* d_ws,
                              size_t ws_size, hipStream_t stream) {
  (void)in_sizes; (void)n_in; (void)out_size; (void)ws_size;
  const float* x     = (const float*)d_in[0];
  const float* W_sx  = (const float*)d_in[1];
  const float* b_sx  = (const float*)d_in[2];
  const float* W_qkv = (const float*)d_in[3];
  const float* rpb   = (const float*)d_in[4];
  const float* sita  = (const float*)d_in[5];
  const float* W_s   = (const float*)d_in[6];
  const float* b_s   = (const float*)d_in[7];
  const float* W_out = (const float*)d_in[8];
  const float* b_out = (const float*)d_in[9];
  const int*   rel_i = (const int*)d_in[10];
  const float* dis   = (const float*)d_in[11];

  float* y     = (float*)d_out;
  float* attn0 = (float*)d_out + (size_t)YELEMS;

  float* fws = (float*)d_ws;
  size_t o = 0;
  float* s1    = fws + o; o += (size_t)NBATCH * DIMF;
  float* xp    = fws + o; o += (size_t)NBATCH * NPATCH * KDP;
  float* Wp    = fws + o; o += (size_t)KDP * QKVN;
  float* Wop   = fws + o; o += (size_t)KO * DIMF;
  float* biasA = fws + o; o += (size_t)NHEADS << 20;
  float* Qm    = fws + o; o += (size_t)BH * NPATCH * DHEAD;
  float* Km    = fws + o; o += (size_t)BH * NPATCH * DHEAD;
  float* Vt    = fws + o; o += (size_t)BH * NPATCH * DHEAD;
  float* out_s = fws + o; o += (size_t)BH * NPATCH * DHEAD;
  float* m0s   = fws + o; o += (size_t)BH * NPATCH;
  float* m1s   = fws + o; o += (size_t)BH * NPATCH;
  float* s0s   = fws + o; o += (size_t)BH * NPATCH;
  float* s2    = fws + o; o += (size_t)BH * DHEAD;
  float* o2    = fws + o; o += (size_t)NBATCH * NPATCH * KO;
  int* idx1 = (int*)(fws + o); o += (size_t)NBATCH * KD;
  int* idx2 = (int*)(fws + o); o += (size_t)BH * KH2;

  scores1_kernel<<<NBATCH, DIMF, 0, stream>>>(x, W_sx, b_sx, s1);
  topk_keep_kernel<<<NBATCH, DIMF, 0, stream>>>(s1, DIMF, KD, idx1);
  gather_xp_kernel<<<(NBATCH * NPATCH * KDP + 255) / 256, 256, 0, stream>>>(
      x, idx1, xp);
  wpad_kernel<<<(KDP * QKVN + 255) / 256, 256, 0, stream>>>(W_qkv, Wp);
  wopad_kernel<<<(KO * DIMF + 255) / 256, 256, 0, stream>>>(W_out, Wop);
  qkv_gemm_kernel<<<(512 * (QKVN / 64)) / 8, 256, 0, stream>>>(xp, Wp, Qm, Km, Vt);
  bias_kernel<<<(1 << 20) / 256, 256, 0, stream>>>(rpb, sita, rel_i, dis, biasA);
  attn_pass1_kernel<<<(BH * 64) / 8, 256, 0, stream>>>(Qm, Km, biasA, m0s, m1s);
  attn_pass2_kernel<<<(BH * 64) / 8, 256, 0, stream>>>(Qm, Km, Vt, biasA, m0s,
                                                       m1s, attn0, s0s, out_s);
  attn0_norm_kernel<<<(BH * NPATCH * NPATCH / 4) / 256, 256, 0, stream>>>(attn0,
                                                                          s0s);
  scores2_kernel<<<BH, DHEAD, 0, stream>>>(out_s, W_s, b_s, s2);
  topk_keep_kernel<<<BH, DHEAD, 0, stream>>>(s2, DHEAD, KH2, idx2);
  gather_out2_kernel<<<(NBATCH * NPATCH * KO) / 256, 256, 0, stream>>>(out_s,
                                                                       idx2, o2);
  out_gemm_kernel<<<(512 * (DIMF / 64)) / 8, 256, 0, stream>>>(o2, Wop, b_out, y);
}